// CPUForgetMult_26611617366440
// MI455X (gfx1250) — compile-verified
//
#include <hip/hip_runtime.h>
#include <hip/hip_bf16.h>

// Problem geometry (fixed by the reference): [seq=2048, batch=64, hidden=512]
#define SEQ     2048
#define CH      32768              // batch * hidden (contiguous fastest dims)
#define BLK_CH  128                // channels per workgroup (one thread each)
#define T_TILE  32                 // timesteps staged per LDS tile
#define TILE_N  (T_TILE * BLK_CH)  // 4096 floats = 16 KiB per array per buffer
#define NBUF    4                  // ring depth (3 tiles in flight)
#define NTILES  (SEQ / T_TILE)     // 64

// ---- CDNA5 async global->LDS path (ASYNCcnt), probed at compile time ------
#if defined(__gfx1250__) && \
    __has_builtin(__builtin_amdgcn_global_load_async_to_lds_b128) && \
    __has_builtin(__builtin_amdgcn_s_wait_asynccnt)
#define USE_ASYNC 1
#else
#define USE_ASYNC 0
#endif

// Builtin params are pointers to 4xi32 clang ext-vectors (confirmed by hipcc
// diagnostic): global source in AS1, LDS destination in AS3.
typedef int v4i __attribute__((ext_vector_type(4)));
typedef v4i __attribute__((address_space(1))) gv4i;   // global
typedef v4i __attribute__((address_space(3))) lv4i;   // LDS

__device__ __forceinline__ void cp16_to_lds(const float* g, float* l) {
#if USE_ASYNC
  // 16B per lane, written straight into LDS by the async engine (ASYNCcnt).
  __builtin_amdgcn_global_load_async_to_lds_b128(
      (gv4i*)(unsigned long long)(const void*)g,
      (lv4i*)(unsigned int)(unsigned long long)(void*)l,
      /*imm offset*/0, /*cpol*/0);
#else
  const float4 v = *(const float4*)g;
  *(float4*)l = v;
#endif
}

// Per wave, one tile = 16 async instructions (8 for f + 8 for x).
__device__ __forceinline__ void wait_async_tiles_pending(int newest_ok) {
#if USE_ASYNC
  if (newest_ok >= 2)      __builtin_amdgcn_s_wait_asynccnt(32);
  else if (newest_ok == 1) __builtin_amdgcn_s_wait_asynccnt(16);
  else                     __builtin_amdgcn_s_wait_asynccnt(0);
#endif
}

__global__ __launch_bounds__(BLK_CH)
void forgetmult_scan_kernel(const float* __restrict__ fg,
                            const float* __restrict__ xg,
                            float* __restrict__ out) {
  __shared__ float sf[NBUF][TILE_N];   // 4 x 16 KiB
  __shared__ float sx[NBUF][TILE_N];   // 4 x 16 KiB  (128 KiB total)

  const int tid  = threadIdx.x;                  // 0..127
  const int c0   = blockIdx.x * BLK_CH;          // first channel of this block
  // copy mapping: 128 threads x 8 issues cover 32 rows x 128 floats (B128 each)
  const int lrow = tid >> 5;                     // 0..3
  const int lcol = (tid & 31) << 2;              // 0,4,...,124

  auto issue_tile = [&](int tile, int buf) {
    const int t0 = tile * T_TILE;
    #pragma unroll
    for (int i = 0; i < 8; ++i) {
      const int row = (i << 2) + lrow;                                 // 0..31
      const long long g = (long long)(t0 + row) * CH + (c0 + lcol);
      const int l = row * BLK_CH + lcol;
      cp16_to_lds(fg + g, &sf[buf][l]);
      cp16_to_lds(xg + g, &sx[buf][l]);
    }
  };

  // Prime the pipeline: 3 tiles in flight.
  issue_tile(0, 0);
  issue_tile(1, 1);
  issue_tile(2, 2);

  float h = 0.0f;                   // hidden state carried in a register
  float* o = out + c0 + tid;        // this thread's channel column

  for (int k = 0; k < NTILES; ++k) {
    const int buf = k & (NBUF - 1);

    // Tiles issued beyond k at this point: min(2, NTILES-1-k).
    if (k + 2 < NTILES)      wait_async_tiles_pending(2);
    else if (k + 1 < NTILES) wait_async_tiles_pending(1);
    else                     wait_async_tiles_pending(0);

    __syncthreads();   // publish tile k; also proves all waves finished tile k-1

    // Buffer (k+3)%NBUF held tile k-1, which everyone is now past -> refill it.
    if (k + 3 < NTILES) issue_tile(k + 3, (k + 3) & (NBUF - 1));

    const int t0 = k * T_TILE;
    #pragma unroll
    for (int r = 0; r < T_TILE; ++r) {
      const float fv = sf[buf][r * BLK_CH + tid];
      const float xv = sx[buf][r * BLK_CH + tid];
      h = fmaf(1.0f - fv, h, fv * xv);    // h = f*x + (1-f)*h
      __builtin_nontemporal_store(h, o + (long long)(t0 + r) * CH);
    }
  }
}

extern "C" void kernel_launch(void* const* d_in, const int* in_sizes, int n_in,
                              void* d_out, int out_size, void* d_ws, size_t ws_size,
                              hipStream_t stream) {
  const float* f = (const float*)d_in[0];   // setup_inputs order: {"f", "x"}
  const float* x = (const float*)d_in[1];
  float* out = (float*)d_out;

  dim3 grid(CH / BLK_CH);   // 256 workgroups
  dim3 block(BLK_CH);       // 128 threads = 4 wave32
  hipLaunchKernelGGL(forgetmult_scan_kernel, grid, block, 0, stream, f, x, out);
}